// DisparityModule_90924457656423
// MI455X (gfx1250) — compile-verified
//
#include <hip/hip_runtime.h>

typedef float v2f __attribute__((ext_vector_type(2)));
typedef float v8f __attribute__((ext_vector_type(8)));

#define N_LAYERS 18
#define N_MATS   (N_LAYERS + 1)   // 18 hidden layers + bias head (Wb)

// Cross-lane gather via LDS permute hardware (no LDS memory touched).
__device__ __forceinline__ float bperm(int idx_bytes, float v) {
    return __int_as_float(__builtin_amdgcn_ds_bpermute(idx_bytes, __float_as_int(v)));
}

// One wave32 runs the whole chain. Each 9x9 (or 3x9) matvec is
// 3 chained V_WMMA_F32_16X16X4_F32 (K padded 9->12, fp32 throughout).
// All weight fragments are loaded UNCONDITIONALLY up front (clamped indices,
// clause-friendly, no exec churn) and garbage-masked via multiply-by-0/1
// masks in the prologue. The per-layer critical path is exactly:
// 3 wmma -> add -> relu -> 6 bpermute. Row outputs are buffered in VGPRs
// and stored in one predicated block at the end.
__global__ __launch_bounds__(32)
void disparity_mlp_wmma(const float* __restrict__ disp,
                        const float* __restrict__ W0,
                        const float* __restrict__ b0,
                        const float* __restrict__ Ws,
                        const float* __restrict__ bs,
                        const float* __restrict__ Wb,
                        const float* __restrict__ bb,
                        float* __restrict__ out)
{
    const int lane  = threadIdx.x;      // 0..31 (wave32)
    const int n     = lane & 15;        // B/D column index
    const int khalf = (lane >> 4) * 2;  // K sub-offset: 0 (lanes 0-15), 2 (lanes 16-31)

    // Branchless 0/1 masks (computed once; applied with v_mul, never exec).
    const float cmask = (n    < 9) ? 1.0f : 0.0f;   // valid B column
    const float rmask = (lane < 9) ? 1.0f : 0.0f;   // valid activation lane

    // ---------- Upfront load of ALL B fragments (unconditional, clamped idx) ----------
    // B[k][n] = W[n][k]. Rows k>=9 multiply zero-padded A entries (harmless);
    // columns n>=9 are zeroed via cmask so no per-layer lane masking is needed.
    v2f bf[N_MATS][3];
    const int nc = (n < 9) ? n : 8;
    #pragma unroll
    for (int i = 0; i < N_LAYERS; ++i) {
        const float* W = Ws + i * 81;   // row-major [9][9]
        #pragma unroll
        for (int c = 0; c < 3; ++c) {
            const int k0 = 4 * c + khalf;
            const int ka = (k0     < 9) ? k0     : 8;
            const int kb = (k0 + 1 < 9) ? k0 + 1 : 8;
            bf[i][c][0] = W[nc * 9 + ka] * cmask;
            bf[i][c][1] = W[nc * 9 + kb] * cmask;
        }
    }
    {   // bias head: B = Wb^T (Wb is [3][9]); only columns n<3 are ever read
        const int nb = (n < 3) ? n : 2;
        #pragma unroll
        for (int c = 0; c < 3; ++c) {
            const int k0 = 4 * c + khalf;
            const int ka = (k0     < 9) ? k0     : 8;
            const int kb = (k0 + 1 < 9) ? k0 + 1 : 8;
            bf[N_LAYERS][c][0] = Wb[nb * 9 + ka];
            bf[N_LAYERS][c][1] = Wb[nb * 9 + kb];
        }
    }

    // Per-layer biases: lane j<9 keeps bs[i][j], zero elsewhere (pre-masked
    // so x = acc[0] + bsv is already 0 on the padded feedback lanes 9..15).
    float bsv[N_LAYERS];
    const int l9 = (lane < 9) ? lane : 8;
    #pragma unroll
    for (int i = 0; i < N_LAYERS; ++i)
        bsv[i] = bs[i * 9 + l9] * rmask;
    const float bbv = bb[(lane < 3) ? lane : 2];

    // ---------- fc0: h[j] = d0*W0[j][0] + d1*W0[j][1] + b0[j] (no ReLU before fc1) ----------
    const float d0 = disp[0], d1 = disp[1];
    float hv = fmaf(d0, W0[l9 * 2 + 0], fmaf(d1, W0[l9 * 2 + 1], b0[l9])) * rmask;

    // bpermute byte indices (lane-constant): A slot (c,r) wants h[4c+khalf+r]
    int pidx0[3], pidx1[3];
    #pragma unroll
    for (int c = 0; c < 3; ++c) {
        pidx0[c] = (4 * c + khalf)     * 4;
        pidx1[c] = (4 * c + khalf + 1) * 4;
    }

    float xrow[N_LAYERS];               // buffered kernel rows (stored at end)
    float xhead = 0.0f;

    // ---------- 19 chained matvecs, fully unrolled ----------
    #pragma unroll
    for (int i = 0; i < N_MATS; ++i) {
        v2f a0, a1, a2;                 // A fragments: every lane holds the row-0 values
        a0[0] = bperm(pidx0[0], hv);  a0[1] = bperm(pidx1[0], hv);
        a1[0] = bperm(pidx0[1], hv);  a1[1] = bperm(pidx1[1], hv);
        a2[0] = bperm(pidx0[2], hv);  a2[1] = bperm(pidx1[2], hv);

        v8f acc = {};
        acc = __builtin_amdgcn_wmma_f32_16x16x4_f32(false, a0, false, bf[i][0],
                                                    (short)0, acc, false, false);
        acc = __builtin_amdgcn_wmma_f32_16x16x4_f32(false, a1, false, bf[i][1],
                                                    (short)0, acc, false, false);
        acc = __builtin_amdgcn_wmma_f32_16x16x4_f32(false, a2, false, bf[i][2],
                                                    (short)0, acc, false, false);

        if (i < N_LAYERS) {
            // B cols >=9 and bsv are pre-zeroed, so x is valid on lanes 0..8
            // and exactly 0 on the padded feedback lanes 9..15. No masking.
            const float x = acc[0] + bsv[i];
            xrow[i] = x;                // kernel row (pre-ReLU, as in reference)
            hv = fmaxf(x, 0.0f);        // ReLU feeds next matvec
        } else {
            xhead = acc[0] + bbv;       // bias head output (lanes 0..2 valid)
        }
    }

    // ---------- single predicated store block (off the compute chain) ----------
    if (lane < 9) {
        #pragma unroll
        for (int i = 0; i < N_LAYERS; ++i)
            out[i * 9 + lane] = xrow[i];
    }
    if (lane < 3)
        out[162 + lane] = xhead;
}

extern "C" void kernel_launch(void* const* d_in, const int* in_sizes, int n_in,
                              void* d_out, int out_size, void* d_ws, size_t ws_size,
                              hipStream_t stream) {
    (void)in_sizes; (void)n_in; (void)d_ws; (void)ws_size; (void)out_size;
    const float* disp = (const float*)d_in[0];  // (1,2)
    const float* W0   = (const float*)d_in[1];  // (9,2)
    const float* b0   = (const float*)d_in[2];  // (9,)
    const float* Ws   = (const float*)d_in[3];  // (18,9,9)
    const float* bs   = (const float*)d_in[4];  // (18,9)
    const float* Wb   = (const float*)d_in[5];  // (3,9)
    const float* bb   = (const float*)d_in[6];  // (3,)
    float* out = (float*)d_out;                 // 162 kernel + 3 bias = 165 floats

    disparity_mlp_wmma<<<1, 32, 0, stream>>>(disp, W0, b0, Ws, bs, Wb, bb, out);
}